// DBTransformerLayer_16724602650673
// MI455X (gfx1250) — compile-verified
//
#include <hip/hip_runtime.h>
#include <math.h>

// ---------------------------------------------------------------------------
// Problem constants (match reference)
// ---------------------------------------------------------------------------
#define NNODE 20000
#define SEQ   8
#define DIM   128
#define NEDGE 100000
#define NHEAD 4
#define DHEAD 32

typedef __attribute__((ext_vector_type(2))) float v2f;
typedef __attribute__((ext_vector_type(8))) float v8f;

// fp32 WMMA: D(16x16,f32) = A(16x4,f32) * B(4x16,f32) + C
// A per lane: m = lane%16, kpair = 2*(lane/16) -> {A[m][k+kp], A[m][k+kp+1]}
// B per lane: n = lane%16, kpair = 2*(lane/16) -> {B[kp][n], B[kp+1][n]}
// C/D vgpr j: row = j + 8*(lane/16), col = lane%16
__device__ __forceinline__ v8f wmma_f32x4(v2f a, v2f b, v8f c) {
    return __builtin_amdgcn_wmma_f32_16x16x4_f32(
        /*neg_a=*/false, a, /*neg_b=*/false, b,
        /*c_mod=*/(short)0, c, /*reuse_a=*/false, /*reuse_b=*/false);
}

// ---------------------------------------------------------------------------
// zero scratch
// ---------------------------------------------------------------------------
__global__ void zero_f32(float* __restrict__ p, size_t n) {
    size_t i = (size_t)blockIdx.x * blockDim.x + threadIdx.x;
    size_t stride = (size_t)gridDim.x * blockDim.x;
    for (; i < n; i += stride) p[i] = 0.0f;
}

// ---------------------------------------------------------------------------
// Fold the source-side projection:  Wsrc = [Wk;Wv] @ Wb   (256 x 128)
//                                   bsrc = [Wk;Wv] @ bb + [bk;bv]
// Win rows: [0:128)=Wq, [128:256)=Wk, [256:384)=Wv  -> row (128+i) spans Wk,Wv
// ---------------------------------------------------------------------------
__global__ __launch_bounds__(128) void fold_src(
    const float* __restrict__ Win, const float* __restrict__ bin,
    const float* __restrict__ Wb,  const float* __restrict__ bb,
    float* __restrict__ Wsrc, float* __restrict__ bsrc)
{
    int i = blockIdx.x;      // 0..255
    int j = threadIdx.x;     // 0..127
    const float* Wrow = Win + (size_t)(128 + i) * DIM;
    float acc = 0.0f;
    for (int k = 0; k < DIM; ++k) acc += Wrow[k] * Wb[(size_t)k * DIM + j];
    Wsrc[(size_t)i * DIM + j] = acc;
    if (j == 0) {
        float b = 0.0f;
        for (int k = 0; k < DIM; ++k) b += Wrow[k] * bb[k];
        bsrc[i] = b + bin[128 + i];
    }
}

// ---------------------------------------------------------------------------
// out[M, NC] = A[M,128] @ W[NC,128]^T + bias[NC]     (fp32 WMMA)
// grid = (M/16, NC/128); block = 256 (8 waves); wave w -> ntile blockIdx.y*8+w
// ---------------------------------------------------------------------------
__global__ __launch_bounds__(256) void gemm_proj(
    const float* __restrict__ A, const float* __restrict__ W,
    const float* __restrict__ bias, float* __restrict__ out, int NC)
{
    int wave = threadIdx.x >> 5;
    int lane = threadIdx.x & 31;
    int half = lane >> 4;       // selects K pair 0/2
    int l16  = lane & 15;
    int mtile = blockIdx.x;
    int ntile = blockIdx.y * 8 + wave;

    int mrow = mtile * 16 + l16;
    int ncol = ntile * 16 + l16;
    const float* Arow = A + (size_t)mrow * DIM;
    const float* Wrow = W + (size_t)ncol * DIM;   // B[k][n] = W[n][k]

    v8f c;
#pragma unroll
    for (int j = 0; j < 8; ++j) c[j] = 0.0f;

#pragma unroll 4
    for (int k = 0; k < DIM; k += 4) {
        v2f a, b;
        a[0] = Arow[k + 2 * half];
        a[1] = Arow[k + 2 * half + 1];
        b[0] = Wrow[k + 2 * half];
        b[1] = Wrow[k + 2 * half + 1];
        c = wmma_f32x4(a, b, c);
    }

    float bv = bias[ncol];
#pragma unroll
    for (int j = 0; j < 8; ++j) {
        int m = mtile * 16 + j + 8 * half;
        out[(size_t)m * NC + ncol] = c[j] + bv;
    }
}

// ---------------------------------------------------------------------------
// Final projection with fused mean + HeteroConv combine:
// out = beta*out + alpha*( (Acc/max(cnt,1)) @ Wo^T + (cnt>0)*bo )
// NC fixed = 128; grid = (M/16, 1); block = 256 (waves cover ntile 0..7)
// ---------------------------------------------------------------------------
__global__ __launch_bounds__(256) void gemm_out(
    const float* __restrict__ Acc, const float* __restrict__ cnt,
    const float* __restrict__ W, const float* __restrict__ bias,
    float* __restrict__ out, float alpha, float beta)
{
    int wave = threadIdx.x >> 5;
    int lane = threadIdx.x & 31;
    int half = lane >> 4;
    int l16  = lane & 15;
    int mtile = blockIdx.x;

    int mrow = mtile * 16 + l16;
    int ncol = wave * 16 + l16;
    float invc = 1.0f / fmaxf(cnt[mrow >> 3], 1.0f);
    const float* Arow = Acc + (size_t)mrow * DIM;
    const float* Wrow = W + (size_t)ncol * DIM;

    v8f c;
#pragma unroll
    for (int j = 0; j < 8; ++j) c[j] = 0.0f;

#pragma unroll 4
    for (int k = 0; k < DIM; k += 4) {
        v2f a, b;
        a[0] = Arow[k + 2 * half] * invc;
        a[1] = Arow[k + 2 * half + 1] * invc;
        b[0] = Wrow[k + 2 * half];
        b[1] = Wrow[k + 2 * half + 1];
        c = wmma_f32x4(a, b, c);
    }

    float bv = bias[ncol];
#pragma unroll
    for (int j = 0; j < 8; ++j) {
        int m = mtile * 16 + j + 8 * half;
        float mask = (cnt[m >> 3] > 0.0f) ? 1.0f : 0.0f;
        float v = alpha * (c[j] + mask * bv);
        size_t idx = (size_t)m * DIM + ncol;
        if (beta != 0.0f) v += beta * out[idx];
        out[idx] = v;
    }
}

// ---------------------------------------------------------------------------
// Per-edge attention. One block (128 thr = 4 waves) per edge, one head/wave.
// Tdst[node] : [SEQ][384] = Q|K|V of dst node;  Tsrc[node] : [SEQ][256] = K|V
// Acc[dst][q][d] += o ;  cnt[dst] += 1
// ---------------------------------------------------------------------------
#define LSTR 132   // padded LDS row stride (floats) to dodge bank conflicts

__global__ __launch_bounds__(128) void edge_attn(
    const float* __restrict__ Tdst, const float* __restrict__ Tsrc,
    const int* __restrict__ ei, float* __restrict__ Acc,
    float* __restrict__ cnt, int E)
{
    __shared__ float Qs[8 * LSTR];
    __shared__ float Ks[16 * LSTR];
    __shared__ float Vs[16 * LSTR];
    __shared__ float P[NHEAD * 8 * 16];

    int e   = blockIdx.x;
    int src = ei[e];
    int dst = ei[E + e];
    int t   = threadIdx.x;
    int row = t >> 4;            // 0..7
    int c16 = (t & 15) * 8;      // 8-float chunk

    const float* db = Tdst + (size_t)dst * (SEQ * 384) + row * 384;
    const float* sb = Tsrc + (size_t)src * (SEQ * 256) + row * 256;

    // Q rows 0..7 (dst cols 0:128)
    *(float4*)&Qs[row * LSTR + c16]     = *(const float4*)&db[c16];
    *(float4*)&Qs[row * LSTR + c16 + 4] = *(const float4*)&db[c16 + 4];
    // K rows 0..7 = dst cols 128:256 ; rows 8..15 = src cols 0:128
    *(float4*)&Ks[row * LSTR + c16]           = *(const float4*)&db[128 + c16];
    *(float4*)&Ks[row * LSTR + c16 + 4]       = *(const float4*)&db[128 + c16 + 4];
    *(float4*)&Ks[(8 + row) * LSTR + c16]     = *(const float4*)&sb[c16];
    *(float4*)&Ks[(8 + row) * LSTR + c16 + 4] = *(const float4*)&sb[c16 + 4];
    // V rows 0..7 = dst cols 256:384 ; rows 8..15 = src cols 128:256
    *(float4*)&Vs[row * LSTR + c16]           = *(const float4*)&db[256 + c16];
    *(float4*)&Vs[row * LSTR + c16 + 4]       = *(const float4*)&db[256 + c16 + 4];
    *(float4*)&Vs[(8 + row) * LSTR + c16]     = *(const float4*)&sb[128 + c16];
    *(float4*)&Vs[(8 + row) * LSTR + c16 + 4] = *(const float4*)&sb[128 + c16 + 4];

    __syncthreads();

    int h    = t >> 5;           // head == wave
    int lane = t & 31;
    int q    = lane >> 2;        // query row 0..7
    int g    = lane & 3;         // group: keys g*4..g*4+3 / dcols g*8..g*8+7
    int hoff = h * DHEAD;

    float qr[DHEAD];
#pragma unroll
    for (int d = 0; d < DHEAD; ++d) qr[d] = Qs[q * LSTR + hoff + d];

    const float scale = 0.1767766952966369f;  // 1/sqrt(32)
    float sc[4];
#pragma unroll
    for (int j = 0; j < 4; ++j) {
        int kk = g * 4 + j;
        float acc = 0.0f;
#pragma unroll
        for (int d = 0; d < DHEAD; ++d) acc += qr[d] * Ks[kk * LSTR + hoff + d];
        sc[j] = acc * scale;
    }
    // softmax over 16 keys (4 lanes x 4 each)
    float mx = fmaxf(fmaxf(sc[0], sc[1]), fmaxf(sc[2], sc[3]));
    mx = fmaxf(mx, __shfl_xor(mx, 1));
    mx = fmaxf(mx, __shfl_xor(mx, 2));
    float es = 0.0f;
#pragma unroll
    for (int j = 0; j < 4; ++j) { sc[j] = __expf(sc[j] - mx); es += sc[j]; }
    es += __shfl_xor(es, 1);
    es += __shfl_xor(es, 2);
    float inv = 1.0f / es;
#pragma unroll
    for (int j = 0; j < 4; ++j) P[h * 128 + q * 16 + g * 4 + j] = sc[j] * inv;

    __syncthreads();

    // o[q][g*8 + j] = sum_kk P[q][kk] * V[kk][hoff + g*8 + j]
    float o[8];
#pragma unroll
    for (int j = 0; j < 8; ++j) o[j] = 0.0f;
#pragma unroll
    for (int kk = 0; kk < 16; ++kk) {
        float p = P[h * 128 + q * 16 + kk];
#pragma unroll
        for (int j = 0; j < 8; ++j) o[j] += p * Vs[kk * LSTR + hoff + g * 8 + j];
    }

    float* acc = Acc + (size_t)dst * (SEQ * DIM) + q * DIM + hoff + g * 8;
#pragma unroll
    for (int j = 0; j < 8; ++j) atomicAdd(&acc[j], o[j]);
    if (t == 0) atomicAdd(&cnt[dst], 1.0f);
}

// ---------------------------------------------------------------------------
// Host-side orchestration
// ---------------------------------------------------------------------------
extern "C" void kernel_launch(void* const* d_in, const int* in_sizes, int n_in,
                              void* d_out, int out_size, void* d_ws, size_t ws_size,
                              hipStream_t stream)
{
    (void)in_sizes; (void)n_in; (void)out_size; (void)ws_size;

    const size_t NS  = (size_t)NNODE * SEQ;    // 160000 rows
    const size_t NSD = NS * DIM;               // 20,480,000

    const float* x_a = (const float*)d_in[0];
    const float* x_b = (const float*)d_in[1];
    const int* ei1 = (const int*)d_in[2];
    const int* ei2 = (const int*)d_in[3];
    const int* ei3 = (const int*)d_in[4];

    float* ws   = (float*)d_ws;
    float* Tdst = ws;                          // NS*384
    float* Tsrc = Tdst + NS * 384;             // NS*256
    float* Acc  = Tsrc + NS * 256;             // NSD
    float* cnt  = Acc + NSD;                   // NNODE
    float* Wsrc = cnt + NNODE;                 // 256*128
    float* bsrc = Wsrc + 256 * 128;            // 256

    float* out_a = (float*)d_out;              // stack index 0
    float* out_b = out_a + NSD;                // stack index 1

    struct Rel {
        const float* xs; const float* xd; const int* ei;
        int pbase; float* outp; float alpha; float beta;
    };
    // out_b = conv_r1 ; out_a = 0.5*(conv_r2 + conv_r3)
    Rel rels[3] = {
        { x_a, x_b, ei1, 5,  out_b, 1.0f, 0.0f },
        { x_b, x_a, ei2, 11, out_a, 0.5f, 0.0f },
        { x_a, x_a, ei3, 17, out_a, 0.5f, 1.0f },
    };

    const int MT = (int)(NS / 16);             // 10000 m-tiles

    for (int r = 0; r < 3; ++r) {
        const float* Wb  = (const float*)d_in[rels[r].pbase + 0];
        const float* bb  = (const float*)d_in[rels[r].pbase + 1];
        const float* Win = (const float*)d_in[rels[r].pbase + 2];
        const float* bin = (const float*)d_in[rels[r].pbase + 3];
        const float* Wo  = (const float*)d_in[rels[r].pbase + 4];
        const float* bo  = (const float*)d_in[rels[r].pbase + 5];

        fold_src<<<256, 128, 0, stream>>>(Win, bin, Wb, bb, Wsrc, bsrc);
        // dst node table: Q|K|V directly from Win ([3D,D] row-blocked)
        gemm_proj<<<dim3(MT, 3), 256, 0, stream>>>(rels[r].xd, Win, bin, Tdst, 384);
        // src node table: folded K|V
        gemm_proj<<<dim3(MT, 2), 256, 0, stream>>>(rels[r].xs, Wsrc, bsrc, Tsrc, 256);
        // zero Acc + cnt (contiguous)
        zero_f32<<<2048, 256, 0, stream>>>(Acc, NSD + NNODE);
        edge_attn<<<NEDGE, 128, 0, stream>>>(Tdst, Tsrc, rels[r].ei, Acc, cnt, NEDGE);
        gemm_out<<<dim3(MT, 1), 256, 0, stream>>>(Acc, cnt, Wo, bo,
                                                  rels[r].outp, rels[r].alpha, rels[r].beta);
    }
}